// GCN_69793218560234
// MI455X (gfx1250) — compile-verified
//
#include <hip/hip_runtime.h>
#include <math.h>

typedef __attribute__((ext_vector_type(2))) float f32x2;
typedef __attribute__((ext_vector_type(8))) float f32x8;

// ---------------- utility kernels ----------------

__global__ void k_zero(float* __restrict__ p, long long n) {
  long long i = (long long)blockIdx.x * blockDim.x + threadIdx.x;
  if (i < n) p[i] = 0.0f;
}

__global__ void k_deg(const int* __restrict__ dst, float* __restrict__ deg, int nE) {
  int e = blockIdx.x * blockDim.x + threadIdx.x;
  if (e < nE) atomicAdd(&deg[dst[e]], 1.0f);
}

__global__ void k_dinv(float* __restrict__ deg, int n) {
  int i = blockIdx.x * blockDim.x + threadIdx.x;
  if (i < n) deg[i] = rsqrtf(deg[i] + 1.0f);  // deg of A+I, in place -> dinv
}

// ---------------- f32 WMMA GEMM: C[M,N] = A[M,K] @ B[K,N], row-major ----------------
// One wave32 per 16x64 output strip (4 N-tiles): the A fragment is loaded once per
// k-step and reused by 4 V_WMMA_F32_16X16X4_F32 ops (4 independent acc chains).
// K, N are compile-time -> fully regular unrolled loop, no scalar tail.
// A 16x4 layout: lanes 0-15 hold (K=0,K=1), lanes 16-31 hold (K=2,K=3), M = lane&15.
// B 4x16 layout: lane = column, VGPR0/1 = K within half (symmetric to A).
// C/D 16x16: VGPR r -> row r (lanes 0-15) / row 8+r (lanes 16-31), col = lane&15.
template <int K, int N>
__global__ void k_gemm_wmma(const float* __restrict__ A, const float* __restrict__ B,
                            float* __restrict__ C, int M) {
  const int m0   = blockIdx.x * 16;
  const int n0   = blockIdx.y * 64;    // 4 tiles of 16 columns
  const int lane = threadIdx.x;        // 0..31, EXEC all ones
  const int half = lane >> 4;          // which K-pair this half-wave owns
  const int lid  = lane & 15;

  const float* ap = A + (size_t)(m0 + lid) * K + (half * 2);
  const float* bp = B + (size_t)(half * 2) * N + n0 + lid;

  f32x8 acc[4];
#pragma unroll
  for (int j = 0; j < 4; ++j) acc[j] = (f32x8){0.f, 0.f, 0.f, 0.f, 0.f, 0.f, 0.f, 0.f};

#pragma unroll 4
  for (int k = 0; k < K; k += 4) {
    f32x2 a;
    a.x = ap[0];
    a.y = ap[1];
    ap += 4;
#pragma unroll
    for (int j = 0; j < 4; ++j) {
      f32x2 b;
      b.x = bp[j * 16];
      b.y = bp[N + j * 16];
      acc[j] = __builtin_amdgcn_wmma_f32_16x16x4_f32(
          /*neg_a=*/false, a, /*neg_b=*/false, b,
          /*c_mod=*/(short)0, acc[j], /*reuse_a=*/false, /*reuse_b=*/false);
    }
    bp += 4 * N;
  }

  float* cp = C + (size_t)(m0 + half * 8) * N + n0 + lid;
#pragma unroll
  for (int j = 0; j < 4; ++j)
#pragma unroll
    for (int r = 0; r < 8; ++r) cp[(size_t)r * N + j * 16] = acc[j][r];
}

// ---------------- edge scatter: agg[dst] += norm * h[src] ----------------
// One thread per (edge, 4-feature chunk). h/agg (<=51.2MB each) stay L2-resident;
// f32 atomics resolve at L2.
__global__ void k_scatter(const float* __restrict__ h, const int* __restrict__ src,
                          const int* __restrict__ dst, const float* __restrict__ dinv,
                          float* __restrict__ agg, int nE, int fshift /*log2(F/4)*/,
                          int F) {
  long long tid = (long long)blockIdx.x * blockDim.x + threadIdx.x;
  long long total = (long long)nE << fshift;
  if (tid >= total) return;
  int e  = (int)(tid >> fshift);
  int fq = (int)(tid & ((1 << fshift) - 1));
  int s = src[e];
  int d = dst[e];
  float norm = dinv[s] * dinv[d];
  const float4 hv = *reinterpret_cast<const float4*>(h + (size_t)s * F + fq * 4);
  float* ap = agg + (size_t)d * F + fq * 4;
  atomicAdd(ap + 0, hv.x * norm);
  atomicAdd(ap + 1, hv.y * norm);
  atomicAdd(ap + 2, hv.z * norm);
  atomicAdd(ap + 3, hv.w * norm);
}

// ---------------- combine: agg += h * dinv^2 + b ; optional ReLU ----------------
__global__ void k_combine(float* __restrict__ agg, const float* __restrict__ h,
                          const float* __restrict__ dinv, const float* __restrict__ bias,
                          long long total, int fshift /*log2(F)*/, int relu) {
  long long idx = (long long)blockIdx.x * blockDim.x + threadIdx.x;
  if (idx >= total) return;
  int i = (int)(idx >> fshift);
  int f = (int)(idx & ((1 << fshift) - 1));
  float di = dinv[i];
  float v = agg[idx] + h[idx] * di * di + bias[f];
  agg[idx] = relu ? fmaxf(v, 0.0f) : v;
}

// ---------------- mean pool accumulation ----------------
__global__ void k_pool(const float* __restrict__ h, const int* __restrict__ batch,
                       float* __restrict__ sums, float* __restrict__ cnts, int n, int F) {
  int i = blockIdx.x;       // node
  int f = threadIdx.x;      // feature (F = 64 = 2 waves)
  if (i >= n) return;
  int g = batch[i];
  atomicAdd(&sums[(size_t)g * F + f], h[(size_t)i * F + f]);
  if (f == 0) atomicAdd(&cnts[g], 1.0f);
}

// ---------------- head: GEMV + sigmoid + BCE mean ----------------
__global__ void k_head(const float* __restrict__ sums, const float* __restrict__ cnts,
                       const float* __restrict__ Wl, const float* __restrict__ bl,
                       const int* __restrict__ y, float* __restrict__ out, int nG, int F) {
  __shared__ float ls[512];
  int g = threadIdx.x;      // 512 threads, one block
  float c = fmaxf(cnts[g], 1.0f);
  float acc = 0.0f;
#pragma unroll 8
  for (int f = 0; f < F; ++f) acc += (sums[(size_t)g * F + f] / c) * Wl[f];
  float logit = acc + bl[0];
  out[g] = 1.0f / (1.0f + expf(-logit));
  float t = (float)y[g];
  ls[g] = fmaxf(logit, 0.0f) - logit * t + log1pf(expf(-fabsf(logit)));
  __syncthreads();
  for (int s = nG >> 1; s > 0; s >>= 1) {
    if (g < s) ls[g] += ls[g + s];
    __syncthreads();
  }
  if (g == 0) out[nG] = ls[0] / (float)nG;
}

// ---------------- launch ----------------

extern "C" void kernel_launch(void* const* d_in, const int* in_sizes, int n_in,
                              void* d_out, int out_size, void* d_ws, size_t ws_size,
                              hipStream_t stream) {
  const float* x   = (const float*)d_in[0];
  const int*   ei  = (const int*)d_in[1];   // [2, E] flat: row0 = src, row1 = dst
  const int*   bat = (const int*)d_in[2];
  const int*   y   = (const int*)d_in[3];
  const float* W1  = (const float*)d_in[4];
  const float* b1  = (const float*)d_in[5];
  const float* W2  = (const float*)d_in[6];
  const float* b2  = (const float*)d_in[7];
  const float* W3  = (const float*)d_in[8];
  const float* b3  = (const float*)d_in[9];
  const float* Wl  = (const float*)d_in[10];
  const float* bl  = (const float*)d_in[11];

  const int nN = 50000, nE = 800000;
  const int F1 = 256, F3 = 64, nG = 512;
  const int* src = ei;
  const int* dst = ei + nE;

  // workspace: two ping-pong node-feature buffers (each sized for F1) + small arrays.
  // Both big buffers (51.2 MB each) fit together in the 192 MB L2.
  float* ws   = (float*)d_ws;
  float* bufA = ws;                          // h  : nN*F1 floats max
  float* bufB = bufA + (size_t)nN * F1;      // agg: nN*F1 floats max
  float* dinv = bufB + (size_t)nN * F1;      // nN floats
  float* sums = dinv + nN;                   // nG*F3 floats
  float* cnts = sums + (size_t)nG * F3;      // nG floats

  auto nb = [](long long n, int bs) { return (unsigned)((n + bs - 1) / bs); };

  // --- degrees of A + I -> dinv ---
  k_zero<<<nb(nN, 256), 256, 0, stream>>>(dinv, nN);
  k_deg<<<nb(nE, 256), 256, 0, stream>>>(dst, dinv, nE);
  k_dinv<<<nb(nN, 256), 256, 0, stream>>>(dinv, nN);

  // --- layer 1: h1 = x @ W1 (50000x128 @ 128x256) ; agg+relu -> bufB ---
  {
    dim3 g(nN / 16, 256 / 64);
    k_gemm_wmma<128, 256><<<g, 32, 0, stream>>>(x, W1, bufA, nN);
    long long tot = (long long)nN * 256;
    k_zero<<<nb(tot, 256), 256, 0, stream>>>(bufB, tot);
    long long et = (long long)nE * (256 / 4);
    k_scatter<<<nb(et, 256), 256, 0, stream>>>(bufA, src, dst, dinv, bufB, nE, 6, 256);
    k_combine<<<nb(tot, 256), 256, 0, stream>>>(bufB, bufA, dinv, b1, tot, 8, 1);
  }

  // --- layer 2: h2 = bufB @ W2 (50000x256 @ 256x128) ; agg+relu -> bufB ---
  {
    dim3 g(nN / 16, 128 / 64);
    k_gemm_wmma<256, 128><<<g, 32, 0, stream>>>(bufB, W2, bufA, nN);
    long long tot = (long long)nN * 128;
    k_zero<<<nb(tot, 256), 256, 0, stream>>>(bufB, tot);
    long long et = (long long)nE * (128 / 4);
    k_scatter<<<nb(et, 256), 256, 0, stream>>>(bufA, src, dst, dinv, bufB, nE, 5, 128);
    k_combine<<<nb(tot, 256), 256, 0, stream>>>(bufB, bufA, dinv, b2, tot, 7, 1);
  }

  // --- layer 3: h3 = bufB @ W3 (50000x128 @ 128x64) ; agg (no relu) -> bufB ---
  {
    dim3 g(nN / 16, 64 / 64);
    k_gemm_wmma<128, 64><<<g, 32, 0, stream>>>(bufB, W3, bufA, nN);
    long long tot = (long long)nN * 64;
    k_zero<<<nb(tot, 256), 256, 0, stream>>>(bufB, tot);
    long long et = (long long)nE * (64 / 4);
    k_scatter<<<nb(et, 256), 256, 0, stream>>>(bufA, src, dst, dinv, bufB, nE, 4, 64);
    k_combine<<<nb(tot, 256), 256, 0, stream>>>(bufB, bufA, dinv, b3, tot, 6, 0);
  }

  // --- global mean pool + head ---
  k_zero<<<nb((long long)nG * F3 + nG, 256), 256, 0, stream>>>(sums, (long long)nG * F3 + nG);
  k_pool<<<nN, F3, 0, stream>>>(bufB, bat, sums, cnts, nN, F3);
  k_head<<<1, nG, 0, stream>>>(sums, cnts, Wl, bl, y, (float*)d_out, nG, F3);
}